// MixtureGaussianSequenceLabeling_88871463289132
// MI455X (gfx1250) — compile-verified
//
#include <hip/hip_runtime.h>
#include <hip/hip_bf16.h>

#define DIMN 32
#define BATCH 32
#define SEQL 64
#define KTOP 2
#define ICN 2
#define TCN 2
#define NLABN 32
#define LOG2PI_F 1.8378770664093453f

typedef __attribute__((ext_vector_type(2))) float v2f;
typedef __attribute__((ext_vector_type(8))) float v8f;

// pointer types for the gfx1250 async Global<->LDS builtins (param type is v4i*)
typedef int v4i __attribute__((vector_size(16)));
typedef __attribute__((address_space(1))) v4i gv4i;   // global
typedef __attribute__((address_space(3))) v4i lv4i;   // LDS

// workspace layout (floats)
constexpr size_t WS_LAM0 = 0;      // TC*64*64 = 8192
constexpr size_t WS_ETA0 = 8192;   // TC*64
constexpr size_t WS_LDTV = 8320;   // TC
constexpr size_t WS_C0   = 8322;   // TC
constexpr size_t WS_STATE = 8328;  // 16B aligned; 2*B*(L+1)*K*ST_STRIDE
// state record: var[0..1023], mu[1024..1055], score[1056], pad -> 1064 (16B multiple)
#define ST_STRIDE 1064

#if __has_builtin(__builtin_amdgcn_global_load_async_to_lds_b128)
#define HAVE_ASYNC_LDS 1
#else
#define HAVE_ASYNC_LDS 0
#endif

__device__ inline void wave_sync() {
  __builtin_amdgcn_wave_barrier();
  asm volatile("s_wait_dscnt 0x0" ::: "memory");
  __builtin_amdgcn_wave_barrier();
}

// 16-byte global -> LDS copy, async on CDNA5 (ASYNCcnt path), fallback float4.
__device__ inline void cp_g2l_16(float* lds_dst, const float* gsrc) {
#if HAVE_ASYNC_LDS
  __builtin_amdgcn_global_load_async_to_lds_b128(
      (gv4i*)gsrc, (lv4i*)lds_dst, 0, 0);
#else
  *(float4*)lds_dst = *(const float4*)gsrc;
#endif
}

// 16-byte LDS -> global copy, async on CDNA5, fallback float4.
__device__ inline void cp_l2g_16(float* gdst, const float* lds_src) {
#if HAVE_ASYNC_LDS && __has_builtin(__builtin_amdgcn_global_store_async_from_lds_b128)
  __builtin_amdgcn_global_store_async_from_lds_b128(
      (gv4i*)gdst, (lv4i*)lds_src, 0, 0);
#else
  *(float4*)gdst = *(const float4*)lds_src;
#endif
}

__device__ inline void async_wait() {
#if HAVE_ASYNC_LDS
#if __has_builtin(__builtin_amdgcn_s_wait_asynccnt)
  __builtin_amdgcn_s_wait_asynccnt(0);
#else
  asm volatile("s_wait_asynccnt 0x0" ::: "memory");
#endif
#endif
}

// ---- WMMA GEMM: C[MxN] = A[MxK] * B[KxN], row-major, M,N mult of 16, K mult of 4.
__device__ inline void mm_wmma(const float* A, int lda, bool transA,
                               const float* B, int ldb,
                               float* C, int ldc, int M, int N, int Kd)
{
  const int lane = threadIdx.x & 31;
  const int half = lane >> 4;
  const int l16  = lane & 15;
  for (int mt = 0; mt < M; mt += 16) {
    for (int nt = 0; nt < N; nt += 16) {
      v8f c = {0.f,0.f,0.f,0.f,0.f,0.f,0.f,0.f};
      for (int k = 0; k < Kd; k += 4) {
        const int ka = k + 2*half;
        v2f a, b;
        if (transA) {
          a.x = A[(ka+0)*lda + mt + l16];
          a.y = A[(ka+1)*lda + mt + l16];
        } else {
          a.x = A[(mt + l16)*lda + ka + 0];
          a.y = A[(mt + l16)*lda + ka + 1];
        }
        b.x = B[(ka+0)*ldb + nt + l16];
        b.y = B[(ka+1)*ldb + nt + l16];
        c = __builtin_amdgcn_wmma_f32_16x16x4_f32(false, a, false, b,
                                                  (short)0, c, false, false);
      }
#pragma unroll
      for (int r = 0; r < 8; ++r)
        C[(mt + r + half*8)*ldc + nt + l16] = c[r];
    }
  }
  wave_sync();
}

// ---- Gauss-Jordan inverse (SPD, no pivoting) in LDS; returns logdet. Destroys M.
__device__ float gj_inv(float* M, float* V, int n)
{
  const int lane = threadIdx.x & 31;
  for (int i = lane; i < n*n; i += 32) V[i] = 0.f;
  wave_sync();
  for (int i = lane; i < n; i += 32) V[i*n + i] = 1.f;
  wave_sync();
  float ld = 0.f;
  for (int p = 0; p < n; ++p) {
    const float piv = M[p*n + p];
    ld += __logf(fabsf(piv) + 1e-37f);
    const float ip = 1.0f / piv;
    for (int c = lane; c < n; c += 32) { M[p*n + c] *= ip; V[p*n + c] *= ip; }
    wave_sync();
    for (int r = 0; r < n; ++r) {
      if (r == p) continue;
      const float f = M[r*n + p];
      for (int c = lane; c < n; c += 32) {
        M[r*n + c] = fmaf(-f, M[p*n + c], M[r*n + c]);
        V[r*n + c] = fmaf(-f, V[p*n + c], V[r*n + c]);
      }
    }
    wave_sync();
  }
  return ld;
}

__device__ inline void matvecg(const float* M, int ld, const float* x, float* y)
{
  const int lane = threadIdx.x & 31;
  float acc = 0.f;
  for (int j = 0; j < 32; ++j) acc += M[lane*ld + j] * x[j];
  y[lane] = acc;
  wave_sync();
}

__device__ inline float dot32(const float* a, const float* b)
{
  const int lane = threadIdx.x & 31;
  float v = a[lane] * b[lane];
  for (int m = 16; m; m >>= 1) v += __shfl_xor(v, m, 32);
  return v;
}

__device__ inline void top2_idx(const float* s, int n, int* outIdx)
{
  int i0 = 0;
  for (int i = 1; i < n; ++i) if (s[i] > s[i0]) i0 = i;
  int i1 = (i0 == 0) ? 1 : 0;
  for (int i = 0; i < n; ++i) { if (i == i0) continue; if (s[i] > s[i1]) i1 = i; }
  outIdx[0] = i0; outIdx[1] = i1;
}

// ---- gaussian_multi with diagonal var1. Returns zeta; muOut(32), varOut(32x32).
__device__ float gmulti_diag(const float* v0, const float* mu0,
                             const float* dvar, const float* mu1,
                             float* bufA, float* bufB,
                             float* muOut, float* varOut, float* vbuf)
{
  const int lane = threadIdx.x & 31;
  for (int i = lane; i < 1024; i += 32) bufA[i] = v0[i];
  wave_sync();
  bufA[lane*32 + lane] += dvar[lane];
  wave_sync();
  const float ldadd = gj_inv(bufA, bufB, 32);
  float *diff = vbuf, *tv = vbuf+32, *tmp = vbuf+64;
  diff[lane] = mu1[lane] - mu0[lane];
  wave_sync();
  matvecg(bufB, 32, diff, tv);
  const float maha = dot32(diff, tv);
  matvecg(v0, 32, tv, tmp);
  muOut[lane] = mu0[lane] + tmp[lane];
  wave_sync();
  mm_wmma(v0, 32, false, bufB, 32, varOut, 32, 32, 32, 32);   // v0 * inv_add
  for (int i = lane; i < 1024; i += 32) varOut[i] *= dvar[i & 31]; // * diag(var1)
  wave_sync();
  return -0.5f * (32.0f * LOG2PI_F + ldadd + maha);
}

// ---- gaussian_multi with full var1.
__device__ float gmulti_full(const float* v0, const float* mu0,
                             const float* v1m, const float* mu1,
                             float* bufA, float* bufB, float* bufC,
                             float* muOut, float* varOut, float* vbuf)
{
  const int lane = threadIdx.x & 31;
  for (int i = lane; i < 1024; i += 32) bufA[i] = v0[i] + v1m[i];
  wave_sync();
  const float ldadd = gj_inv(bufA, bufB, 32);
  float *diff = vbuf, *tv = vbuf+32, *tmp = vbuf+64;
  diff[lane] = mu1[lane] - mu0[lane];
  wave_sync();
  matvecg(bufB, 32, diff, tv);
  const float maha = dot32(diff, tv);
  matvecg(v0, 32, tv, tmp);
  muOut[lane] = mu0[lane] + tmp[lane];
  wave_sync();
  mm_wmma(v0, 32, false, bufB, 32, bufC, 32, 32, 32, 32);
  mm_wmma(bufC, 32, false, v1m, 32, varOut, 32, 32, 32, 32);
  return -0.5f * (32.0f * LOG2PI_F + ldadd + maha);
}

// ---- score-only gaussian_multi with diagonal var1.
__device__ float gscore_diag(const float* v0, const float* mu0,
                             const float* dvar, const float* mu1,
                             float* bufA, float* bufB, float* vbuf)
{
  const int lane = threadIdx.x & 31;
  for (int i = lane; i < 1024; i += 32) bufA[i] = v0[i];
  wave_sync();
  bufA[lane*32 + lane] += dvar[lane];
  wave_sync();
  const float ldadd = gj_inv(bufA, bufB, 32);
  float *diff = vbuf, *tv = vbuf+32;
  diff[lane] = mu1[lane] - mu0[lane];
  wave_sync();
  matvecg(bufB, 32, diff, tv);
  const float maha = dot32(diff, tv);
  return -0.5f * (32.0f * LOG2PI_F + ldadd + maha);
}

// ---- gaussian_multi_integral via block-Schur on lam_new = lam0 + pad(lam1).
__device__ float integral_schur(const float* lam0, const float* eta0,
                                float ld0, float c0s,
                                const float* lam1, const float* eta1,
                                float ldv1, float c1, bool fwd,
                                float* bufA, float* bufB, float* bufC,
                                float* muOut, float* varOut, float* vbuf)
{
  const int lane = threadIdx.x & 31;
  const int off = fwd ? 0 : 32;
  const int oth = 32 - off;
  for (int i = lane; i < 1024; i += 32) {
    const int r = i >> 5, c = i & 31;
    bufA[i] = lam0[(off + r)*64 + off + c] + lam1[i];     // P
  }
  wave_sync();
  const float ldP = gj_inv(bufA, bufB, 32);               // bufB = P^-1
  mm_wmma(bufB, 32, false, lam0 + off*64 + oth, 64, bufC, 32, 32, 32, 32); // P^-1 Q
  mm_wmma(lam0 + oth*64 + off, 64, false, bufC, 32, bufA, 32, 32, 32, 32); // R P^-1 Q
  for (int i = lane; i < 1024; i += 32) {
    const int r = i >> 5, c = i & 31;
    bufA[i] = lam0[(oth + r)*64 + oth + c] - bufA[i];     // Schur S
  }
  wave_sync();
  const float ldS = gj_inv(bufA, varOut, 32);             // varOut = S^-1 = out var
  float *e_p = vbuf, *e_o = vbuf+32, *t1 = vbuf+64, *t2 = vbuf+96,
        *x_o = vbuf+128, *x_p = vbuf+160;
  e_p[lane] = eta0[off + lane] + eta1[lane];
  e_o[lane] = eta0[oth + lane];
  wave_sync();
  matvecg(bufB, 32, e_p, t1);                             // P^-1 e_p
  matvecg(lam0 + oth*64 + off, 64, t1, t2);               // R P^-1 e_p
  t2[lane] = e_o[lane] - t2[lane];
  wave_sync();
  matvecg(varOut, 32, t2, x_o);                           // x_o = out mu block
  matvecg(lam0 + off*64 + oth, 64, x_o, t2);              // Q x_o
  t2[lane] = e_p[lane] - t2[lane];
  wave_sync();
  matvecg(bufB, 32, t2, x_p);
  const float med = dot32(x_p, e_p) + dot32(x_o, e_o);    // mu_new . eta_new
  muOut[lane] = x_o[lane];
  wave_sync();
  return -0.5f * (32.0f * LOG2PI_F + ld0 + ldv1 + (ldP + ldS) + c0s + c1 - med);
}

// ============================= kernel 1: transition precompute =================
__global__ __launch_bounds__(64)
void k_precompute(const float* tmu, const float* tcho, float* ws)
{
  __shared__ __align__(16) float sm[16384];
  const int wv = threadIdx.x >> 5;
  const int lane = threadIdx.x & 31;
  const int tc = wv;
  float* M = sm + wv*8192;
  float* V = M + 4096;
  const float* cho = tcho + tc*64*64;
  mm_wmma(cho, 64, true, cho, 64, M, 64, 64, 64, 64);       // tv = cho^T cho
  const float ld = gj_inv(M, V, 64);                        // V = lam0, ld = logdet(tv)
  float* lam0 = ws + WS_LAM0 + (size_t)tc*4096;
  for (int i = lane; i < 4096; i += 32) lam0[i] = V[i];
  const float* tm = tmu + tc*64;
  float a0 = 0.f, a1 = 0.f;
  for (int j = 0; j < 64; ++j) { a0 += V[lane*64+j]*tm[j]; a1 += V[(lane+32)*64+j]*tm[j]; }
  ws[WS_ETA0 + tc*64 + lane]      = a0;
  ws[WS_ETA0 + tc*64 + 32 + lane] = a1;
  float part = tm[lane]*a0 + tm[lane+32]*a1;
  for (int m = 16; m; m >>= 1) part += __shfl_xor(part, m, 32);
  if (lane == 0) { ws[WS_LDTV + tc] = ld; ws[WS_C0 + tc] = part; }
}

// ============================= kernel 2: fwd/bwd HMM scan ======================
constexpr int SM_LAM0=0, SM_ETA0=8192, SM_LDTV=8320, SM_C0=8322,
              SM_PSC=8324, SM_PMU=8328, SM_PVAR=8392,
              SM_CSC=10440, SM_CMU=10448, SM_CVAR=10704,
              SM_INMU=18896, SM_INVAR=18960,
              SM_WBUF=19024, SM_WVEC=35408, SM2_TOTAL=36432;

__device__ inline float* state_ptr(float* ws, int dir, int b, int t, int k)
{
  return ws + WS_STATE + (size_t)((((dir*BATCH + b)*(SEQL+1) + t)*KTOP) + k) * ST_STRIDE;
}

__device__ void commit_state(float* sm, const int* sIdx, float* ws, int dir, int b, int t)
{
  const int tid = threadIdx.x;
  for (int k = 0; k < KTOP; ++k) {
    const int src = sIdx[k];
    float* g = state_ptr(ws, dir, b, t, k);
    if (tid == 0) { const float v = sm[SM_CSC+src]; sm[SM_PSC+k] = v; g[1056] = v; }
    for (int i = tid; i < 32;   i += 128) sm[SM_PMU+k*32+i]   = sm[SM_CMU+src*32+i];
    for (int i = tid; i < 1024; i += 128) sm[SM_PVAR+k*1024+i] = sm[SM_CVAR+src*1024+i];
    // async LDS -> global streaming of the selected hypothesis
    for (int c = tid; c < 256; c += 128) cp_l2g_16(g + c*4,        sm + SM_CVAR + src*1024 + c*4);
    for (int c = tid; c < 8;   c += 128) cp_l2g_16(g + 1024 + c*4, sm + SM_CMU  + src*32  + c*4);
  }
  async_wait();   // cand buffers are rewritten next step; stores must drain
}

__global__ __launch_bounds__(128)
void k_scan(const int* sent, const float* mu_emb, const float* cho_emb, float* ws)
{
  __shared__ __align__(16) float sm[SM2_TOTAL];
  __shared__ int sIdx[2];
  const int tid = threadIdx.x, lane = tid & 31, wv = tid >> 5;
  const int b = blockIdx.x & 31;
  const int dir = blockIdx.x >> 5;          // 0 = forward, 1 = backward
  const bool fwd = (dir == 0);

  // async bulk load of lam0/eta0 into LDS
  for (int c = tid; c < 2048; c += 128) cp_g2l_16(sm + SM_LAM0 + c*4, ws + WS_LAM0 + c*4);
  for (int c = tid; c < 32;   c += 128) cp_g2l_16(sm + SM_ETA0 + c*4, ws + WS_ETA0 + c*4);
  if (tid < 2) { sm[SM_LDTV+tid] = ws[WS_LDTV+tid]; sm[SM_C0+tid] = ws[WS_C0+tid]; }
  async_wait();
  __syncthreads();

  float* wb = sm + SM_WBUF + wv*4096;
  float *bufA = wb, *bufB = wb+1024, *bufC = wb+2048, *bufD = wb+3072;
  float* vb = sm + SM_WVEC + wv*256;

  // -------- t = 0: integral(tm, 0, tv, I) -> TC candidates
  if (wv < TCN) {
    const int tc = wv;
    for (int i = lane; i < 1024; i += 32) bufD[i] = 0.f;
    wave_sync();
    bufD[lane*32 + lane] = 1.f;               // lam1 = I
    float* e1 = vb + 192;
    e1[lane] = 0.f;                            // eta1 = 0
    wave_sync();
    const float z = integral_schur(sm+SM_LAM0+tc*4096, sm+SM_ETA0+tc*64,
                                   sm[SM_LDTV+tc], sm[SM_C0+tc],
                                   bufD, e1, 0.f, 0.f, fwd,
                                   bufA, bufB, bufC,
                                   sm+SM_CMU+tc*32, sm+SM_CVAR+tc*1024, vb);
    if (lane == 0) sm[SM_CSC + tc] = z;
  }
  __syncthreads();
  if (tid == 0) top2_idx(sm+SM_CSC, TCN, sIdx);
  __syncthreads();
  commit_state(sm, sIdx, ws, dir, b, 0);
  __syncthreads();

  // -------- scan steps
  for (int t = 1; t <= SEQL; ++t) {
    const int tokpos = fwd ? (t - 1) : (SEQL - t);
    const int token = sent[b*SEQL + tokpos];
    for (int j = tid; j < 64; j += 128) {
      sm[SM_INMU + j] = mu_emb[(size_t)token*64 + j];
      const float c = cho_emb[(size_t)token*64 + j];
      sm[SM_INVAR + j] = c * c;
    }
    if (t < SEQL && tid == 0) {               // prefetch next embedding row
      const int npos = fwd ? t : (SEQL - t - 1);
      const int ntok = sent[b*SEQL + npos];
      __builtin_prefetch(&mu_emb[(size_t)ntok*64], 0, 0);
      __builtin_prefetch(&cho_emb[(size_t)ntok*64], 0, 0);
    }
    __syncthreads();

    const int kh = wv >> 1, ic = wv & 1;
    const float ps = sm[SM_PSC + kh];
    const float* pm = sm + SM_PMU + kh*32;
    const float* pv = sm + SM_PVAR + kh*1024;
    float* muM = vb + 224;
    const float ms = gmulti_diag(pv, pm, sm+SM_INVAR+ic*32, sm+SM_INMU+ic*32,
                                 bufA, bufB, muM, bufC, vb);   // bufC = mv
    for (int i = lane; i < 1024; i += 32) bufA[i] = bufC[i];
    wave_sync();
    const float ldmv = gj_inv(bufA, bufD, 32);                 // bufD = lam1
    float* e1 = vb + 192;
    matvecg(bufD, 32, muM, e1);                                // eta1
    const float c1 = dot32(muM, e1);
    for (int tc = 0; tc < TCN; ++tc) {
      const int cand = wv*TCN + tc;
      const float z = integral_schur(sm+SM_LAM0+tc*4096, sm+SM_ETA0+tc*64,
                                     sm[SM_LDTV+tc], sm[SM_C0+tc],
                                     bufD, e1, ldmv, c1, fwd,
                                     bufA, bufB, bufC,
                                     sm+SM_CMU+cand*32, sm+SM_CVAR+cand*1024, vb);
      if (lane == 0) sm[SM_CSC + cand] = ps + ms + z;
    }
    __syncthreads();
    if (tid == 0) top2_idx(sm+SM_CSC, KTOP*ICN*TCN, sIdx);
    __syncthreads();
    commit_state(sm, sIdx, ws, dir, b, t);
    __syncthreads();
  }
}

// ============================= kernel 3: merge + emission ======================
constexpr int S3_FS=0, S3_FMU=4, S3_FVAR=68,
              S3_BS=2116, S3_BMU=2120, S3_BVAR=2184,
              S3_INMU=4232, S3_INVAR=4296,
              S3_S1=4360, S3_M1=4364, S3_V1=4492,
              S3_CSC=8588, S3_CMU=8596, S3_CVAR=8852,
              S3_ES=17044, S3_EM=17048, S3_EV=17112,
              S3_OMU=19160, S3_OVAR=20184, S3_SC=21208,
              S3_WBUF=21272, S3_WVEC=33560, S3_TOTAL=34584;

__global__ __launch_bounds__(128)
void k_emit(const int* sent, const float* mu_emb, const float* cho_emb,
            const float* omu, const float* ocho, float* ws, float* out)
{
  __shared__ __align__(16) float sm[S3_TOTAL];
  __shared__ int sIdx[2];
  const int tid = threadIdx.x, lane = tid & 31, wv = tid >> 5;
  const int b = blockIdx.x >> 6;
  const int l = blockIdx.x & 63;

  const float* gf = state_ptr(ws, 0, b, l, 0);
  const float* gb = state_ptr(ws, 1, b, SEQL-1-l, 0);
  for (int k = 0; k < KTOP; ++k) {
    if (tid == 0) { sm[S3_FS+k] = gf[k*ST_STRIDE + 1056]; sm[S3_BS+k] = gb[k*ST_STRIDE + 1056]; }
    for (int c = tid; c < 256; c += 128) {
      cp_g2l_16(sm + S3_FVAR + k*1024 + c*4, (float*)gf + k*ST_STRIDE + c*4);
      cp_g2l_16(sm + S3_BVAR + k*1024 + c*4, (float*)gb + k*ST_STRIDE + c*4);
    }
    for (int c = tid; c < 8; c += 128) {
      cp_g2l_16(sm + S3_FMU + k*32 + c*4, (float*)gf + k*ST_STRIDE + 1024 + c*4);
      cp_g2l_16(sm + S3_BMU + k*32 + c*4, (float*)gb + k*ST_STRIDE + 1024 + c*4);
    }
  }
  const int token = sent[b*SEQL + l];
  for (int c = tid; c < 16;  c += 128) cp_g2l_16(sm + S3_INMU + c*4, mu_emb + (size_t)token*64 + c*4);
  for (int c = tid; c < 256; c += 128) cp_g2l_16(sm + S3_OMU + c*4, omu + c*4);
  for (int j = tid; j < 64; j += 128) {
    const float cch = cho_emb[(size_t)token*64 + j];
    sm[S3_INVAR+j] = cch * cch;
  }
  for (int i = tid; i < 1024; i += 128) {
    const float cch = ocho[i]; sm[S3_OVAR+i] = cch * cch;
  }
  async_wait();
  __syncthreads();

  float* wb = sm + S3_WBUF + wv*3072;
  float *bufA = wb, *bufB = wb+1024, *bufC = wb+2048;
  float* vb = sm + S3_WVEC + wv*256;

  // stage A: fuse forward state with input component (wave = (kh, ic))
  {
    const int kh = wv >> 1, ic = wv & 1;
    const float s = gmulti_diag(sm+S3_FVAR+kh*1024, sm+S3_FMU+kh*32,
                                sm+S3_INVAR+ic*32, sm+S3_INMU+ic*32,
                                bufA, bufB, sm+S3_M1+wv*32, sm+S3_V1+wv*1024, vb);
    if (lane == 0) sm[S3_S1+wv] = s + sm[S3_FS+kh];
  }
  __syncthreads();
  // stage B: fuse with backward hypotheses (full x full)
  for (int j = 0; j < KTOP; ++j) {
    const int cand = wv*KTOP + j;
    const float s2 = gmulti_full(sm+S3_V1+wv*1024, sm+S3_M1+wv*32,
                                 sm+S3_BVAR+j*1024, sm+S3_BMU+j*32,
                                 bufA, bufB, bufC,
                                 sm+S3_CMU+cand*32, sm+S3_CVAR+cand*1024, vb);
    if (lane == 0) sm[S3_CSC+cand] = sm[S3_S1+wv] + s2 + sm[S3_BS+j];
  }
  __syncthreads();
  if (tid == 0) top2_idx(sm+S3_CSC, 8, sIdx);
  __syncthreads();
  for (int k = 0; k < KTOP; ++k) {
    const int src = sIdx[k];
    if (tid == 0) sm[S3_ES+k] = sm[S3_CSC+src];
    for (int i = tid; i < 32;   i += 128) sm[S3_EM+k*32+i]   = sm[S3_CMU+src*32+i];
    for (int i = tid; i < 1024; i += 128) sm[S3_EV+k*1024+i] = sm[S3_CVAR+src*1024+i];
  }
  __syncthreads();
  // stage D: label scoring, wave handles 16 of 64 (k,label) pairs
  for (int q = 0; q < 16; ++q) {
    const int idx = wv*16 + q;
    const int k = idx >> 5, lab = idx & 31;
    const float z = gscore_diag(sm+S3_EV+k*1024, sm+S3_EM+k*32,
                                sm+S3_OVAR+lab*32, sm+S3_OMU+lab*32,
                                bufA, bufB, vb);
    if (lane == 0) sm[S3_SC + idx] = z + sm[S3_ES+k];
  }
  __syncthreads();
  if (tid < NLABN) {
    const float a = sm[S3_SC + tid], c = sm[S3_SC + 32 + tid];
    const float m = fmaxf(a, c);
    out[((size_t)b*SEQL + l)*NLABN + tid] = m + __logf(__expf(a - m) + __expf(c - m));
  }
}

extern "C" void kernel_launch(void* const* d_in, const int* in_sizes, int n_in,
                              void* d_out, int out_size, void* d_ws, size_t ws_size,
                              hipStream_t stream)
{
  (void)in_sizes; (void)n_in; (void)out_size; (void)ws_size;
  const int*   sent   = (const int*)d_in[0];
  const float* muEmb  = (const float*)d_in[1];
  const float* choEmb = (const float*)d_in[2];
  const float* tmu    = (const float*)d_in[3];
  const float* tcho   = (const float*)d_in[4];
  const float* omu    = (const float*)d_in[5];
  const float* ocho   = (const float*)d_in[6];
  float* ws  = (float*)d_ws;
  float* out = (float*)d_out;

  k_precompute<<<1, 64, 0, stream>>>(tmu, tcho, ws);
  k_scan<<<BATCH*2, 128, 0, stream>>>(sent, muEmb, choEmb, ws);
  k_emit<<<BATCH*SEQL, 128, 0, stream>>>(sent, muEmb, choEmb, omu, ocho, ws, out);
}